// RecurrentRGCN_74431783240007
// MI455X (gfx1250) — compile-verified
//
#include <hip/hip_runtime.h>
#include <math.h>

// RecurrentRGCN layer for MI455X (gfx1250, wave32).
//
// Restructure: edge MLP input concat is split across W_edge rows so the
// per-edge GEMM becomes three dense node/type-level GEMMs + an elementwise
// gather pass. All dense GEMMs run on V_WMMA_F32_16X16X4_F32 (full fp32,
// matches reference numerics; ~131 GFLOP total).
//
// GEMM shape: each wave computes a 32x128 strip (2 M-tiles x 8 N-tiles of
// 16x16): per K-step, 10 x b64 fragment loads feed 16 WMMAs. K/lda/ldb/ldc
// are compile-time constants so all fragment loads are one base pointer +
// immediate offsets (no per-tile pointer registers -> no spills), and
// __launch_bounds__(128,1) lets the 128 accumulator VGPRs live in registers.
// Weight panels are pre-transposed so every B fragment is a contiguous b64.
//
// Workspace requirement: 2561*N + 400*256 + 4*256*256 floats  ~= 1.03 GB.

typedef __attribute__((ext_vector_type(2))) float v2f;
typedef __attribute__((ext_vector_type(8))) float v8f;

__device__ __forceinline__ float waveSum(float x) {
#pragma unroll
  for (int off = 16; off > 0; off >>= 1) x += __shfl_xor(x, off, 32);
  return x;
}

__device__ __forceinline__ float sigmoidf(float x) {
  return 1.0f / (1.0f + __expf(-x));
}

// ---------------------------------------------------------------------------
// fp32 WMMA GEMM: C[M x Ncols] = A[M x 256] * Bt^T. Bt layout: (k,n) at
// Bt[n*256 + k]. Wave tile = 32 rows x 128 cols; M must be a multiple of 32.
// A-fragment per ISA 7.12.2 (32-bit A 16x4): lane L holds row m=L%16,
// VGPR0 = K = k+2*(L/16), VGPR1 = K+1. B symmetric with n=L%16. C/D: VGPR r
// holds row r+8*(L/16), col L%16.
// ---------------------------------------------------------------------------
template <int LDC>
__global__ __launch_bounds__(128, 1) void gemm_wmma_f32(
    const float* __restrict__ A, const float* __restrict__ Bt,
    float* __restrict__ C, int Mwaves) {
  constexpr int K = 256, LDA = 256, LDB = 256;
  const int lane = threadIdx.x & 31;
  const int wave = threadIdx.x >> 5;
  const int wtile = blockIdx.x * 4 + wave;   // 32-row strip index
  if (wtile >= Mwaves) return;
  const int hi = lane >> 4;   // 0: lanes 0-15, 1: lanes 16-31
  const int lo = lane & 15;
  const int m0 = wtile * 32;
  const int n0 = blockIdx.y * 128;
  // Single per-lane base pointers; all other addressing is immediate offsets.
  const float* abase = A + (size_t)(m0 + lo) * LDA + 2 * hi;
  const float* bbase = Bt + (size_t)(n0 + lo) * LDB + 2 * hi;

  v8f acc[2][8] = {};
  for (int k = 0; k < K; k += 4) {
    v2f a0, a1;
    {
      const float2 t0 = *(const float2*)(abase + k);            // +0
      const float2 t1 = *(const float2*)(abase + k + 16 * LDA); // +16KB imm
      a0.x = t0.x; a0.y = t0.y;
      a1.x = t1.x; a1.y = t1.y;
    }
#pragma unroll
    for (int t = 0; t < 8; ++t) {
      v2f b;
      {
        const float2 tb = *(const float2*)(bbase + k + t * 16 * LDB);
        b.x = tb.x; b.y = tb.y;
      }
      acc[0][t] = __builtin_amdgcn_wmma_f32_16x16x4_f32(
          false, a0, false, b, (short)0, acc[0][t], false, false);
      acc[1][t] = __builtin_amdgcn_wmma_f32_16x16x4_f32(
          false, a1, false, b, (short)0, acc[1][t], false, false);
    }
  }
  // Epilogue: lane-dependent base + constant offsets.
  float* cbase = C + (size_t)(m0 + 8 * hi) * LDC + n0 + lo;
#pragma unroll
  for (int mt = 0; mt < 2; ++mt) {
#pragma unroll
    for (int t = 0; t < 8; ++t) {
#pragma unroll
      for (int r = 0; r < 8; ++r) {
        cbase[(size_t)(mt * 16 + r) * LDC + t * 16] = acc[mt][t][r];
      }
    }
  }
}

// ---------------------------------------------------------------------------
__global__ void zero_f(float* __restrict__ p, size_t n) {
  size_t i = (size_t)blockIdx.x * blockDim.x + threadIdx.x;
  if (i < n) p[i] = 0.0f;
}

// out[n*256 + k] = in[k*256 + n]  (256x256 panel; writes coalesced)
__global__ void transpose256(const float* __restrict__ in,
                             float* __restrict__ out) {
  const int n = blockIdx.x;
  const int k = threadIdx.x;
  out[(size_t)n * 256 + k] = in[(size_t)k * 256 + n];
}

// T3[r, c] = sum_k type_emb[r, k] * W_edge[512 + k, c]   (400 x 64 x 256, tiny)
__global__ void type_mlp(const float* __restrict__ type_emb,
                         const float* __restrict__ W_edge,
                         float* __restrict__ T3) {
  const int r = blockIdx.x;
  const int c = threadIdx.x;
  float s = 0.0f;
#pragma unroll 8
  for (int k = 0; k < 64; ++k)
    s += type_emb[r * 64 + k] * W_edge[(size_t)(512 + k) * 256 + c];
  T3[(size_t)r * 256 + c] = s;
}

// One wave per edge: pre = P1[src]+P2[dst]+T3[et]+b; LN; ReLU; atomic scatter.
__global__ void edge_scatter(const float* __restrict__ P1,
                             const float* __restrict__ P2,
                             const float* __restrict__ T3,
                             const float* __restrict__ b_edge,
                             const float* __restrict__ g,
                             const float* __restrict__ bb,
                             const int* __restrict__ src,
                             const int* __restrict__ dst,
                             const int* __restrict__ et,
                             float* __restrict__ agg,
                             float* __restrict__ cnt, int E) {
  const int gid = blockIdx.x * blockDim.x + threadIdx.x;
  const int e = gid >> 5, lane = gid & 31;
  if (e >= E) return;
  const int s = src[e], d = dst[e], t = et[e];
  const float* p1 = P1 + (size_t)s * 256;
  const float* p2 = P2 + (size_t)d * 256;
  const float* p3 = T3 + (size_t)t * 256;
  float v[8];
  float sum = 0.0f, sq = 0.0f;
#pragma unroll
  for (int j = 0; j < 8; ++j) {
    const int c = j * 32 + lane;
    const float x = p1[c] + p2[c] + p3[c] + b_edge[c];
    v[j] = x; sum += x; sq += x * x;
  }
  sum = waveSum(sum);
  sq = waveSum(sq);
  const float mean = sum * (1.0f / 256.0f);
  const float var = sq * (1.0f / 256.0f) - mean * mean;
  const float rstd = rsqrtf(var + 1e-5f);
  float* ad = agg + (size_t)d * 256;
#pragma unroll
  for (int j = 0; j < 8; ++j) {
    const int c = j * 32 + lane;
    float y = (v[j] - mean) * rstd * g[c] + bb[c];
    y = fmaxf(y, 0.0f);
    unsafeAtomicAdd(ad + c, y);
  }
  if (lane == 0) unsafeAtomicAdd(cnt + d, 1.0f);
}

// edge_enh = agg / max(cnt, 1)   (in place on agg)
__global__ void finalize_mean(float* __restrict__ agg,
                              const float* __restrict__ cnt) {
  const size_t i = (size_t)blockIdx.x * blockDim.x + threadIdx.x;
  agg[i] = agg[i] / fmaxf(cnt[i >> 8], 1.0f);
}

// One wave per node: gate = sigmoid(LN(G1+G2+b)); x = 0.2*node + 0.8*fused.
__global__ void fuse_gate(const float* __restrict__ node,
                          const float* __restrict__ enh,
                          const float* __restrict__ G1,
                          const float* __restrict__ G2,
                          const float* __restrict__ b_gate,
                          const float* __restrict__ g,
                          const float* __restrict__ bb,
                          float* __restrict__ X) {
  const int gid = blockIdx.x * blockDim.x + threadIdx.x;
  const int i = gid >> 5, lane = gid & 31;
  float pre[8], ne[8], eh[8];
  float sum = 0.0f, sq = 0.0f;
#pragma unroll
  for (int j = 0; j < 8; ++j) {
    const int c = j * 32 + lane;
    const size_t o = (size_t)i * 256 + c;
    const float p = G1[o] + G2[o] + b_gate[c];
    pre[j] = p; ne[j] = node[o]; eh[j] = enh[o];
    sum += p; sq += p * p;
  }
  sum = waveSum(sum);
  sq = waveSum(sq);
  const float mean = sum * (1.0f / 256.0f);
  const float var = sq * (1.0f / 256.0f) - mean * mean;
  const float rstd = rsqrtf(var + 1e-5f);
#pragma unroll
  for (int j = 0; j < 8; ++j) {
    const int c = j * 32 + lane;
    const size_t o = (size_t)i * 256 + c;
    const float gate = sigmoidf((pre[j] - mean) * rstd * g[c] + bb[c]);
    const float fused = gate * ne[j] + (1.0f - gate) * eh[j];
    X[o] = 0.2f * ne[j] + 0.8f * fused;
  }
}

// One wave per node: GRU gates from gi/gh, LN, +0.3*h_prev residual.
__global__ void gru_out(const float* __restrict__ GI,
                        const float* __restrict__ GH,
                        const float* __restrict__ hprev,
                        const float* __restrict__ b_ih,
                        const float* __restrict__ b_hh,
                        const float* __restrict__ g,
                        const float* __restrict__ bb,
                        float* __restrict__ out) {
  const int gid = blockIdx.x * blockDim.x + threadIdx.x;
  const int i = gid >> 5, lane = gid & 31;
  const size_t o3 = (size_t)i * 768;
  float hv[8], hp[8];
  float sum = 0.0f, sq = 0.0f;
#pragma unroll
  for (int j = 0; j < 8; ++j) {
    const int c = j * 32 + lane;
    const float r = sigmoidf(GI[o3 + c] + b_ih[c] + GH[o3 + c] + b_hh[c]);
    const float z = sigmoidf(GI[o3 + 256 + c] + b_ih[256 + c] +
                             GH[o3 + 256 + c] + b_hh[256 + c]);
    const float n = tanhf(GI[o3 + 512 + c] + b_ih[512 + c] +
                          r * (GH[o3 + 512 + c] + b_hh[512 + c]));
    const float hpv = hprev[(size_t)i * 256 + c];
    const float h = (1.0f - z) * n + z * hpv;
    hv[j] = h; hp[j] = hpv; sum += h; sq += h * h;
  }
  sum = waveSum(sum);
  sq = waveSum(sq);
  const float mean = sum * (1.0f / 256.0f);
  const float var = sq * (1.0f / 256.0f) - mean * mean;
  const float rstd = rsqrtf(var + 1e-5f);
#pragma unroll
  for (int j = 0; j < 8; ++j) {
    const int c = j * 32 + lane;
    out[(size_t)i * 256 + c] =
        (hv[j] - mean) * rstd * g[c] + bb[c] + 0.3f * hp[j];
  }
}

// ---------------------------------------------------------------------------
extern "C" void kernel_launch(void* const* d_in, const int* in_sizes, int n_in,
                              void* d_out, int out_size, void* d_ws,
                              size_t ws_size, hipStream_t stream) {
  const float* node_emb = (const float*)d_in[0];
  const float* h_prev   = (const float*)d_in[1];
  const float* type_emb = (const float*)d_in[2];
  const float* W_edge   = (const float*)d_in[3];
  const float* b_edge   = (const float*)d_in[4];
  const float* ln_eg    = (const float*)d_in[5];
  const float* ln_eb    = (const float*)d_in[6];
  const float* W_gate   = (const float*)d_in[7];
  const float* b_gate   = (const float*)d_in[8];
  const float* ln_gg    = (const float*)d_in[9];
  const float* ln_gb    = (const float*)d_in[10];
  const float* W_ih     = (const float*)d_in[11];
  const float* W_hh     = (const float*)d_in[12];
  const float* b_ih     = (const float*)d_in[13];
  const float* b_hh     = (const float*)d_in[14];
  const float* ln_rg    = (const float*)d_in[15];
  const float* ln_rb    = (const float*)d_in[16];
  const int* src = (const int*)d_in[17];
  const int* dst = (const int*)d_in[18];
  const int* et  = (const int*)d_in[19];
  float* out = (float*)d_out;

  const int Nn = in_sizes[0] / 256;  // 100000 (divisible by 32)
  const int E  = in_sizes[17];       // 200000

  // Workspace layout (floats); P1/P2 are reused for G2/X after edge pass.
  float* ws = (float*)d_ws;
  const size_t NF = (size_t)Nn;
  float* P1  = ws;                 // 256N : node_emb @ W_edge[0:256]
  float* P2  = P1 + 256 * NF;      // 256N : node_emb @ W_edge[256:512]
  float* AGG = P2 + 256 * NF;      // 256N : scatter sum -> edge_enh
  float* G1  = AGG + 256 * NF;     // 256N : node_emb @ W_gate[0:256]
  float* GH  = G1 + 256 * NF;      // 768N : h_prev @ W_hh^T
  float* GI  = GH + 768 * NF;      // 768N : x @ W_ih^T
  float* CNT = GI + 768 * NF;      // N    : in-degree counts
  float* T3  = CNT + NF;           // 400*256 : type_emb @ W_edge[512:576]
  float* WT1 = T3 + 400 * 256;     // 256x256 : W_edge[0:256]^T
  float* WT2 = WT1 + 256 * 256;    // 256x256 : W_edge[256:512]^T
  float* WTg1 = WT2 + 256 * 256;   // 256x256 : W_gate[0:256]^T
  float* WTg2 = WTg1 + 256 * 256;  // 256x256 : W_gate[256:512]^T
  float* G2  = P1;                 // reuse: edge_enh @ W_gate[256:512]
  float* X   = P2;                 // reuse: fused input to GRU

  const int Mwaves = Nn / 32;  // 3125 exact (32-row strips)
  const dim3 gemmBlk(128, 1, 1);                         // 4 waves/block
  const dim3 g256((unsigned)(Mwaves + 3) / 4, 2, 1);     // Ncols = 256
  const dim3 g768((unsigned)(Mwaves + 3) / 4, 6, 1);     // Ncols = 768

  // 0) init scatter buffers (ws is poisoned by harness); transpose weights
  {
    const size_t n = 256 * NF;
    zero_f<<<(unsigned)((n + 255) / 256), 256, 0, stream>>>(AGG, n);
    zero_f<<<(unsigned)((NF + 255) / 256), 256, 0, stream>>>(CNT, NF);
  }
  transpose256<<<256, 256, 0, stream>>>(W_edge, WT1);
  transpose256<<<256, 256, 0, stream>>>(W_edge + 256 * 256, WT2);
  transpose256<<<256, 256, 0, stream>>>(W_gate, WTg1);
  transpose256<<<256, 256, 0, stream>>>(W_gate + 256 * 256, WTg2);
  // 1) tiny type-embedding GEMM
  type_mlp<<<400, 256, 0, stream>>>(type_emb, W_edge, T3);
  // 2) dense WMMA GEMMs that don't depend on the scatter
  gemm_wmma_f32<256><<<g256, gemmBlk, 0, stream>>>(node_emb, WT1,  P1, Mwaves);
  gemm_wmma_f32<256><<<g256, gemmBlk, 0, stream>>>(node_emb, WT2,  P2, Mwaves);
  gemm_wmma_f32<256><<<g256, gemmBlk, 0, stream>>>(node_emb, WTg1, G1, Mwaves);
  gemm_wmma_f32<768><<<g768, gemmBlk, 0, stream>>>(h_prev,   W_hh, GH, Mwaves);
  // 3) per-edge LN/ReLU + scatter-sum (one wave per edge)
  edge_scatter<<<(unsigned)((E * 32 + 255) / 256), 256, 0, stream>>>(
      P1, P2, T3, b_edge, ln_eg, ln_eb, src, dst, et, AGG, CNT, E);
  // 4) scatter-mean
  finalize_mean<<<(unsigned)((256 * NF) / 256), 256, 0, stream>>>(AGG, CNT);
  // 5) gate GEMM on edge_enh, then fusion -> X
  gemm_wmma_f32<256><<<g256, gemmBlk, 0, stream>>>(AGG, WTg2, G2, Mwaves);
  fuse_gate<<<(unsigned)((Nn * 32 + 255) / 256), 256, 0, stream>>>(
      node_emb, AGG, G1, G2, b_gate, ln_gg, ln_gb, X);
  // 6) GRU input GEMM, then gates + LN + residual -> out
  gemm_wmma_f32<768><<<g768, gemmBlk, 0, stream>>>(X, W_ih, GI, Mwaves);
  gru_out<<<(unsigned)((Nn * 32 + 255) / 256), 256, 0, stream>>>(
      GI, GH, h_prev, b_ih, b_hh, ln_rg, ln_rb, out);
}